// TGA_29326036697877
// MI455X (gfx1250) — compile-verified
//
#include <hip/hip_runtime.h>

typedef float v2f __attribute__((ext_vector_type(2)));
typedef float v8f __attribute__((ext_vector_type(8)));

#define NN 10000
#define NE 50000

static __device__ inline v8f wmma4(v2f a, v2f b, v8f c) {
  // D = A(16x4 f32) * B(4x16 f32) + C(16x16 f32)
  return __builtin_amdgcn_wmma_f32_16x16x4_f32(false, a, false, b, (short)0, c, false, false);
}

// Load eight 16x4 A-operands (K=32) from an LDS 16x32 tile.
static __device__ inline void loadA8(const float* lds, int nl, int koff, v2f a[8]) {
#pragma unroll
  for (int kk = 0; kk < 8; ++kk) {
    int kb = kk * 4 + koff;
    a[kk].x = lds[nl * 32 + kb];
    a[kk].y = lds[nl * 32 + kb + 1];
  }
}

// C[16x32] = A[16x32] @ W[32x32] using hoisted A-operands.
static __device__ inline void gemm32p(const v2f a[8], const float* __restrict__ W,
                                      int nl, int koff, v8f& c0, v8f& c1) {
  c0 = {}; c1 = {};
#pragma unroll
  for (int kk = 0; kk < 8; ++kk) {
    int kA = kk * 4 + koff;
    v2f b0, b1;
    b0.x = W[kA * 32 + nl];       b0.y = W[(kA + 1) * 32 + nl];
    b1.x = W[kA * 32 + 16 + nl];  b1.y = W[(kA + 1) * 32 + 16 + nl];
    c0 = wmma4(a[kk], b0, c0);
    c1 = wmma4(a[kk], b1, c1);
  }
}

static __device__ inline void bias_relu_to_lds(float* lds, v8f c0, v8f c1,
                                               const float* __restrict__ Bv,
                                               int nl, int half, bool relu) {
  float b0 = Bv[nl], b1 = Bv[nl + 16];
#pragma unroll
  for (int r = 0; r < 8; ++r) {
    float y0 = c0[r] + b0, y1 = c1[r] + b1;
    if (relu) { y0 = fmaxf(y0, 0.f); y1 = fmaxf(y1, 0.f); }
    lds[(half * 8 + r) * 32 + nl]      = y0;
    lds[(half * 8 + r) * 32 + 16 + nl] = y1;
  }
}

// ---------------------------------------------------------------------------
// Y[M x 32] = act(X[M x K] @ W[K x 32] + b); one wave per 16-row tile.
// K compile-time; pad rows handled with branch-free index clamps (A pads = 0).
// ---------------------------------------------------------------------------
template <int K, bool RELU>
__global__ __launch_bounds__(32) void lin32_t(const float* __restrict__ X, int ldx,
                                              const float* __restrict__ W,
                                              const float* __restrict__ Bv,
                                              float* __restrict__ Y) {
  constexpr int Kp = (K + 3) & ~3;
  __shared__ float xs[16 * Kp];
  const int lane = threadIdx.x;
  const int row0 = blockIdx.x * 16;
  for (int idx = lane; idx < 16 * Kp; idx += 32) {
    int r = idx / Kp, c = idx - r * Kp;
    int cc = (Kp != K && c >= K) ? (K - 1) : c;
    float v = X[(size_t)(row0 + r) * ldx + cc];
    xs[idx] = (Kp != K && c >= K) ? 0.f : v;
  }
  __syncthreads();
  const int half = lane >> 4, nl = lane & 15, koff = half * 2;
  v8f acc0 = {}, acc1 = {};
#pragma unroll
  for (int k0 = 0; k0 < Kp; k0 += 4) {
    int kA = k0 + koff, kB = kA + 1;
    v2f a; a.x = xs[nl * Kp + kA]; a.y = xs[nl * Kp + kB];
    int kAc = kA, kBc = kB;
    if constexpr (Kp != K) {  // clamp; A pad columns are zero so value is ignored
      kAc = (kA < K) ? kA : K - 1;
      kBc = (kB < K) ? kB : K - 1;
    }
    v2f b0, b1;
    b0.x = W[kAc * 32 + nl];       b0.y = W[kBc * 32 + nl];
    b1.x = W[kAc * 32 + 16 + nl];  b1.y = W[kBc * 32 + 16 + nl];
    acc0 = wmma4(a, b0, acc0);
    acc1 = wmma4(a, b1, acc1);
  }
  float bias0 = Bv[nl], bias1 = Bv[nl + 16];
#pragma unroll
  for (int r = 0; r < 8; ++r) {
    float y0 = acc0[r] + bias0, y1 = acc1[r] + bias1;
    if (RELU) { y0 = fmaxf(y0, 0.f); y1 = fmaxf(y1, 0.f); }
    size_t rw = (size_t)(row0 + half * 8 + r) * 32;
    Y[rw + nl] = y0;
    Y[rw + 16 + nl] = y1;
  }
}

// ---------------------------------------------------------------------------
// Fused hypernetwork bilinear, TWO 16-row tiles per wave (B operands reused):
//   U[e,o] = sum_i x[e,i]*(z@Ww)[i*32+o] + (x@wbMat)[e,o] + (z@Bw)[e,o] + bb[o]
// EDGE=1: x = concat(Xg[dst,6:6+NF], Xg[src,6:6+NF], EA[e,:]), z = CN[src]
// EDGE=0: x = Xg[row], z = CN[row]  (DIN==32)
// ---------------------------------------------------------------------------
template <int DIN, int NF, int EF, bool EDGE>
__global__ __launch_bounds__(32) void mi2_k(const float* __restrict__ Xg, int ldx,
                                            const float* __restrict__ EA,
                                            const int* __restrict__ SRC,
                                            const int* __restrict__ DST,
                                            const float* __restrict__ CN,
                                            const float* __restrict__ Ww,
                                            const float* __restrict__ wb,
                                            const float* __restrict__ Bw,
                                            const float* __restrict__ bb,
                                            float* __restrict__ U, int M) {
  constexpr int DINP = (DIN + 3) & ~3;
  __shared__ float xs[32 * DINP];
  __shared__ float zs[32 * 32];
  __shared__ int sidx[32];
  __shared__ int didx[32];
  const int lane = threadIdx.x;
  const int row0 = blockIdx.x * 32;

  if (EDGE) {
    int e = row0 + lane;
    int ec = (e < M) ? e : (M - 1);
    sidx[lane] = SRC[ec];
    didx[lane] = DST[ec];
    __syncthreads();
    for (int t = 0; t < 32; ++t)
      zs[t * 32 + lane] = CN[(size_t)sidx[t] * 32 + lane];
    for (int t = 0; t < 32; ++t) {
      int ee = row0 + t; ee = (ee < M) ? ee : (M - 1);
      for (int c = lane; c < DINP; c += 32) {
        float v = 0.f;
        if (c < NF)            v = Xg[(size_t)didx[t] * ldx + 6 + c];
        else if (c < 2 * NF)   v = Xg[(size_t)sidx[t] * ldx + 6 + c - NF];
        else if (c < DIN)      v = EA[(size_t)ee * EF + (c - 2 * NF)];
        xs[t * DINP + c] = v;
      }
    }
  } else {
    for (int t = 0; t < 32; ++t) {
      int r = row0 + t; int rc = (r < M) ? r : (M - 1);
      zs[t * 32 + lane] = CN[(size_t)rc * 32 + lane];
      xs[t * 32 + lane] = Xg[(size_t)rc * ldx + lane];
    }
  }
  __syncthreads();

  const int half = lane >> 4, nl = lane & 15, koff = half * 2;
  v2f aZ0[8], aZ1[8];
  loadA8(zs, nl, koff, aZ0);            // rows 0..15
  loadA8(zs + 16 * 32, nl, koff, aZ1);  // rows 16..31

  v8f p00 = {}, p01 = {}, p10 = {}, p11 = {};  // [tile][o-half]

  // z @ Bw  (B reused by both tiles)
#pragma unroll
  for (int kk = 0; kk < 8; ++kk) {
    int kA = kk * 4 + koff;
    v2f b0, b1;
    b0.x = Bw[kA * 32 + nl];       b0.y = Bw[(kA + 1) * 32 + nl];
    b1.x = Bw[kA * 32 + 16 + nl];  b1.y = Bw[(kA + 1) * 32 + 16 + nl];
    p00 = wmma4(aZ0[kk], b0, p00);  p01 = wmma4(aZ0[kk], b1, p01);
    p10 = wmma4(aZ1[kk], b0, p10);  p11 = wmma4(aZ1[kk], b1, p11);
  }

  // x @ wbMat (wb viewed as [DIN x 32]); pad rows index-clamped (A pads zero)
#pragma unroll
  for (int k0 = 0; k0 < DINP; k0 += 4) {
    int kA = k0 + koff, kB = kA + 1;
    v2f a0, a1;
    a0.x = xs[nl * DINP + kA];            a0.y = xs[nl * DINP + kB];
    a1.x = xs[(16 + nl) * DINP + kA];     a1.y = xs[(16 + nl) * DINP + kB];
    int kAc = kA, kBc = kB;
    if constexpr (DINP != DIN) {
      kAc = (kA < DIN) ? kA : DIN - 1;
      kBc = (kB < DIN) ? kB : DIN - 1;
    }
    v2f b0, b1;
    b0.x = wb[kAc * 32 + nl];       b0.y = wb[kBc * 32 + nl];
    b1.x = wb[kAc * 32 + 16 + nl];  b1.y = wb[kBc * 32 + 16 + nl];
    p00 = wmma4(a0, b0, p00);  p01 = wmma4(a0, b1, p01);
    p10 = wmma4(a1, b0, p10);  p11 = wmma4(a1, b1, p11);
  }

  // Main bilinear: Wp sub-tiles produced by WMMA, consumed immediately.
  constexpr int LDW = DIN * 32;
  for (int i = 0; i < DIN; ++i) {
    float s0[8], s1[8];
#pragma unroll
    for (int r = 0; r < 8; ++r) {
      s0[r] = xs[(half * 8 + r) * DINP + i];
      s1[r] = xs[(16 + half * 8 + r) * DINP + i];
    }
    int nb0 = i * 32 + nl;
    v8f w0 = {}, w1 = {};
#pragma unroll
    for (int kk = 0; kk < 8; ++kk) {
      int kA = kk * 4 + koff;
      v2f b; b.x = Ww[kA * LDW + nb0]; b.y = Ww[(kA + 1) * LDW + nb0];
      w0 = wmma4(aZ0[kk], b, w0);
      w1 = wmma4(aZ1[kk], b, w1);
    }
#pragma unroll
    for (int r = 0; r < 8; ++r) { p00[r] += s0[r] * w0[r]; p10[r] += s1[r] * w1[r]; }

    int nb1 = nb0 + 16;
    v8f v0 = {}, v1 = {};
#pragma unroll
    for (int kk = 0; kk < 8; ++kk) {
      int kA = kk * 4 + koff;
      v2f b; b.x = Ww[kA * LDW + nb1]; b.y = Ww[(kA + 1) * LDW + nb1];
      v0 = wmma4(aZ0[kk], b, v0);
      v1 = wmma4(aZ1[kk], b, v1);
    }
#pragma unroll
    for (int r = 0; r < 8; ++r) { p01[r] += s0[r] * v0[r]; p11[r] += s1[r] * v1[r]; }
  }

  float b0v = bb[nl], b1v = bb[nl + 16];
#pragma unroll
  for (int r = 0; r < 8; ++r) {
    int row = row0 + half * 8 + r;
    if (row < M) {
      U[(size_t)row * 32 + nl]      = p00[r] + b0v;
      U[(size_t)row * 32 + 16 + nl] = p01[r] + b1v;
    }
    int row2 = row + 16;
    if (row2 < M) {
      U[(size_t)row2 * 32 + nl]      = p10[r] + b0v;
      U[(size_t)row2 * 32 + 16 + nl] = p11[r] + b1v;
    }
  }
}

// ---------------------------------------------------------------------------
// Fused edge-MLP (4x 32->32) + attn-MLP (3x 32->32 + 32->1) per 16-edge tile.
// u_j staged once in LDS; C->A relayout via LDS between layers.
// ---------------------------------------------------------------------------
__global__ __launch_bounds__(32) void mlp_ea_k(
    const float* __restrict__ uj,
    const float* __restrict__ eW0, const float* __restrict__ eB0,
    const float* __restrict__ eW1, const float* __restrict__ eB1,
    const float* __restrict__ eW2, const float* __restrict__ eB2,
    const float* __restrict__ eW3, const float* __restrict__ eB3,
    const float* __restrict__ aW0, const float* __restrict__ aB0,
    const float* __restrict__ aW1, const float* __restrict__ aB1,
    const float* __restrict__ aW2, const float* __restrict__ aB2,
    const float* __restrict__ aW3, const float* __restrict__ aB3,
    float* __restrict__ hjp, float* __restrict__ zj) {
  __shared__ float su[16 * 32];
  __shared__ float r0[16 * 32];
  __shared__ float r1[16 * 32];
  const int lane = threadIdx.x;
  const int row0 = blockIdx.x * 16;
  for (int t = 0; t < 16; ++t)
    su[t * 32 + lane] = uj[(size_t)(row0 + t) * 32 + lane];
  __syncthreads();
  const int half = lane >> 4, nl = lane & 15, koff = half * 2;
  v2f aU[8], aT[8];
  loadA8(su, nl, koff, aU);
  v8f c0, c1;

  // edge MLP
  gemm32p(aU, eW0, nl, koff, c0, c1);
  bias_relu_to_lds(r0, c0, c1, eB0, nl, half, true);
  __syncthreads();
  loadA8(r0, nl, koff, aT);
  gemm32p(aT, eW1, nl, koff, c0, c1);
  bias_relu_to_lds(r1, c0, c1, eB1, nl, half, true);
  __syncthreads();
  loadA8(r1, nl, koff, aT);
  gemm32p(aT, eW2, nl, koff, c0, c1);
  bias_relu_to_lds(r0, c0, c1, eB2, nl, half, true);
  __syncthreads();
  loadA8(r0, nl, koff, aT);
  gemm32p(aT, eW3, nl, koff, c0, c1);
  {
    float b0 = eB3[nl], b1 = eB3[nl + 16];
#pragma unroll
    for (int r = 0; r < 8; ++r) {
      size_t rw = (size_t)(row0 + half * 8 + r) * 32;
      hjp[rw + nl]      = c0[r] + b0;
      hjp[rw + 16 + nl] = c1[r] + b1;
    }
  }

  // attn MLP
  gemm32p(aU, aW0, nl, koff, c0, c1);
  bias_relu_to_lds(r1, c0, c1, aB0, nl, half, true);
  __syncthreads();
  loadA8(r1, nl, koff, aT);
  gemm32p(aT, aW1, nl, koff, c0, c1);
  bias_relu_to_lds(r0, c0, c1, aB1, nl, half, true);
  __syncthreads();
  loadA8(r0, nl, koff, aT);
  gemm32p(aT, aW2, nl, koff, c0, c1);
  bias_relu_to_lds(r1, c0, c1, aB2, nl, half, true);
  __syncthreads();
  if (lane < 16) {
    float s = 0.f;
#pragma unroll
    for (int k = 0; k < 32; ++k) s += r1[lane * 32 + k] * aW3[k];
    zj[row0 + lane] = s + aB3[0];
  }
}

// --------------------------- small VALU kernels ----------------------------
__global__ void init_k(float* p, int n, float v) {
  int i = blockIdx.x * blockDim.x + threadIdx.x;
  if (i < n) p[i] = v;
}

__global__ void segmax_k(const float* __restrict__ z, const int* __restrict__ dst,
                         float* __restrict__ m, int E) {
  int e = blockIdx.x * blockDim.x + threadIdx.x;
  if (e >= E) return;
  float v = z[e];
  int* addr = (int*)&m[dst[e]];
  int cur = *addr;
  while (v > __int_as_float(cur)) {
    int old = atomicCAS(addr, cur, __float_as_int(v));
    if (old == cur) break;
    cur = old;
  }
}

__global__ void segexp_k(const float* __restrict__ z, const int* __restrict__ dst,
                         const float* __restrict__ m, float* __restrict__ s,
                         float* __restrict__ alf, int E) {
  int e = blockIdx.x * blockDim.x + threadIdx.x;
  if (e >= E) return;
  int d = dst[e];
  float ez = __expf(z[e] - m[d]);
  alf[e] = ez;
  atomicAdd(&s[d], ez);
}

__global__ void segsc_k(const float* __restrict__ h, const float* __restrict__ alf,
                        const float* __restrict__ s, const int* __restrict__ dst,
                        float* __restrict__ emb, int E) {
  int t = blockIdx.x * blockDim.x + threadIdx.x;
  if (t >= E * 32) return;
  int e = t >> 5, o = t & 31;
  int d = dst[e];
  float a = alf[e] / (s[d] + 1e-16f);
  atomicAdd(&emb[(size_t)d * 32 + o], h[t] * a);
}

__global__ void cat_k(const float* __restrict__ x0, int ldx, int nf,
                      const float* __restrict__ ui, float* __restrict__ out, int n) {
  int t = blockIdx.x * blockDim.x + threadIdx.x;
  int cw = nf + 32;
  if (t >= n * cw) return;
  int r = t / cw, c = t - r * cw;
  out[t] = (c < nf) ? x0[(size_t)r * ldx + 6 + c] : ui[(size_t)r * 32 + c - nf];
}

__global__ void xnew_k(const float* __restrict__ xg, int ldx,
                       const float* __restrict__ hipv, float* __restrict__ xn, int n) {
  int t = blockIdx.x * blockDim.x + threadIdx.x;
  if (t >= n * 38) return;
  int r = t / 38, c = t - r * 38;
  xn[t] = (c < 6) ? xg[(size_t)r * ldx + c] : hipv[(size_t)r * 32 + c - 6];
}

__global__ void eic_k(const int* __restrict__ e1, const int* __restrict__ e2,
                      int* __restrict__ o, int E) {
  int e = blockIdx.x * blockDim.x + threadIdx.x;
  if (e >= E) return;
  o[e]         = e1[e];
  o[E + e]     = e2[e];
  o[2 * E + e] = e1[E + e];
  o[3 * E + e] = e2[E + e];
}

// ---------------------------------------------------------------------------
extern "C" void kernel_launch(void* const* d_in, const int* in_sizes, int n_in,
                              void* d_out, int out_size, void* d_ws, size_t ws_size,
                              hipStream_t stream) {
  const int N = NN, E = NE;
  const float* x1  = (const float*)d_in[0];
  const float* x2  = (const float*)d_in[1];
  const float* ea1 = (const float*)d_in[2];
  const float* ea2 = (const float*)d_in[3];
  // params: 44 leaves per layer (jax sorted-dict flatten), l1 at 4, l2 at 48
  const int* ei1 = (const int*)d_in[92];
  const int* ei2 = (const int*)d_in[93];
  auto P = [&](int i) { return (const float*)d_in[i]; };

  float* w = (float*)d_ws;
  size_t off = 0;
  auto A = [&](size_t n) { float* p = w + off; off += n; return p; };
  float* hjp1 = A((size_t)E * 32);
  float* hjp2 = A((size_t)E * 32);
  float* uj   = A((size_t)E * 32);
  float* t0   = A((size_t)E * 32);
  float* t1   = A((size_t)E * 32);
  float* cn   = A((size_t)N * 32);
  float* emb  = A((size_t)N * 32);
  float* ci   = A((size_t)N * 32);
  float* ui   = A((size_t)N * 32);
  float* hipb = A((size_t)N * 32);
  float* catb = A((size_t)N * 64);
  float* xn1  = A((size_t)N * 38);
  float* xn2  = A((size_t)N * 38);
  float* zj   = A((size_t)E);
  float* alf  = A((size_t)E);
  float* segm = A((size_t)N);
  float* segs = A((size_t)N);
  (void)in_sizes; (void)n_in; (void)out_size; (void)ws_size;

  // layer leaf offsets (sorted keys: attn, edge, etype, mi_edge, mi_node, node, ntype)
  auto run_layer = [&](int pb, const float* xa, const float* xb, int xdim, int nf,
                       const float* eaa, const float* eab, int ef,
                       float* hjpa, float* hjpb, float* hipOut) {
    init_k<<<(N * 32 + 255) / 256, 256, 0, stream>>>(emb, N * 32, 0.f);
    const float* xg[2]  = {xa, xb};
    const float* eag[2] = {eaa, eab};
    const int*   eis[2] = {ei1, ei2};
    float*      hjps[2] = {hjpa, hjpb};
    for (int g = 0; g < 2; ++g) {
      const int* src = eis[g];
      const int* dst = eis[g] + E;
      // etype MLP (per node, gathered per edge inside mi2_k)
      lin32_t<6, true><<<N / 16, 32, 0, stream>>>(xg[g], xdim, P(pb + 16), P(pb + 17), t0);
      lin32_t<32, true><<<N / 16, 32, 0, stream>>>(t0, 32, P(pb + 18), P(pb + 19), t1);
      lin32_t<32, false><<<N / 16, 32, 0, stream>>>(t1, 32, P(pb + 20), P(pb + 21), cn);
      // hypernetwork edge bilinear (2 tiles per wave)
      if (nf == 11) {
        mi2_k<23, 11, 1, true><<<(E + 31) / 32, 32, 0, stream>>>(
            xg[g], xdim, eag[g], src, dst, cn,
            P(pb + 23), P(pb + 25), P(pb + 22), P(pb + 24), uj, E);
      } else {
        mi2_k<96, 32, 32, true><<<(E + 31) / 32, 32, 0, stream>>>(
            xg[g], xdim, eag[g], src, dst, cn,
            P(pb + 23), P(pb + 25), P(pb + 22), P(pb + 24), uj, E);
      }
      // fused edge MLP -> h_jp and attn MLP -> z_j
      mlp_ea_k<<<E / 16, 32, 0, stream>>>(
          uj,
          P(pb + 8), P(pb + 9), P(pb + 10), P(pb + 11),
          P(pb + 12), P(pb + 13), P(pb + 14), P(pb + 15),
          P(pb + 0), P(pb + 1), P(pb + 2), P(pb + 3),
          P(pb + 4), P(pb + 5), P(pb + 6), P(pb + 7),
          hjps[g], zj);
      // segment softmax + weighted segment sum into emb (merged across graphs)
      init_k<<<(N + 255) / 256, 256, 0, stream>>>(segm, N, -3.0e38f);
      init_k<<<(N + 255) / 256, 256, 0, stream>>>(segs, N, 0.f);
      segmax_k<<<(E + 255) / 256, 256, 0, stream>>>(zj, dst, segm, E);
      segexp_k<<<(E + 255) / 256, 256, 0, stream>>>(zj, dst, segm, segs, alf, E);
      segsc_k<<<(E * 32 + 255) / 256, 256, 0, stream>>>(hjps[g], alf, segs, dst, emb, E);
    }
    // node path: ntype MLP on graphs[0] features
    lin32_t<6, true><<<N / 16, 32, 0, stream>>>(xa, xdim, P(pb + 38), P(pb + 39), t0);
    lin32_t<32, true><<<N / 16, 32, 0, stream>>>(t0, 32, P(pb + 40), P(pb + 41), t1);
    lin32_t<32, false><<<N / 16, 32, 0, stream>>>(t1, 32, P(pb + 42), P(pb + 43), ci);
    // hypernetwork node bilinear
    mi2_k<32, 0, 0, false><<<(N + 31) / 32, 32, 0, stream>>>(
        emb, 32, nullptr, nullptr, nullptr, ci,
        P(pb + 27), P(pb + 29), P(pb + 26), P(pb + 28), ui, N);
    // node MLP
    int cw = nf + 32;
    cat_k<<<(N * cw + 255) / 256, 256, 0, stream>>>(xa, xdim, nf, ui, catb, N);
    if (nf == 11)
      lin32_t<43, true><<<N / 16, 32, 0, stream>>>(catb, cw, P(pb + 30), P(pb + 31), t0);
    else
      lin32_t<64, true><<<N / 16, 32, 0, stream>>>(catb, cw, P(pb + 30), P(pb + 31), t0);
    lin32_t<32, true><<<N / 16, 32, 0, stream>>>(t0, 32, P(pb + 32), P(pb + 33), t1);
    lin32_t<32, true><<<N / 16, 32, 0, stream>>>(t1, 32, P(pb + 34), P(pb + 35), t0);
    lin32_t<32, false><<<N / 16, 32, 0, stream>>>(t0, 32, P(pb + 36), P(pb + 37), hipOut);
  };

  float* outF = (float*)d_out;
  // layer 1: nf=11, ef=1, xdim=17
  run_layer(4, x1, x2, 17, 11, ea1, ea2, 1, hjp1, hjp2, hipb);
  xnew_k<<<(N * 38 + 255) / 256, 256, 0, stream>>>(x1, 17, hipb, xn1, N);
  xnew_k<<<(N * 38 + 255) / 256, 256, 0, stream>>>(x2, 17, hipb, xn2, N);

  // output layout: h_node [N*32] | ei_m [4E ints] | ea_m [2*E*32]
  float* eaOut1 = outF + (size_t)N * 32 + (size_t)4 * E;
  float* eaOut2 = eaOut1 + (size_t)E * 32;
  // layer 2: nf=32, ef=32, xdim=38; h_jp -> ea_m region, h_ip -> h_node region
  run_layer(48, xn1, xn2, 38, 32, hjp1, hjp2, 32, eaOut1, eaOut2, outF);

  eic_k<<<(E + 255) / 256, 256, 0, stream>>>(ei1, ei2, (int*)d_out + (size_t)N * 32, E);
}